// Xd_73813307949719
// MI455X (gfx1250) — compile-verified
//
#include <hip/hip_runtime.h>

// X_d qudit shift (dim=8, wire index=3, 8 wires, N = 8^8 = 16,777,216).
// M is a permutation matrix, so the whole op is:
//     out[a, j, b] = x[a, (8-j)%8, b]   (a<512, j<8, b<4096)
// packed as interleaved complex64 (re, im) into d_out.
//
// Pure data movement: 128 MB in + 128 MB out  ->  ~11 us at 23.3 TB/s HBM.
// Strategy: b128 loads/stores, perfect coalescing (permutation is contiguous
// at 16 KB granularity), CDNA5 non-temporal cache hints (th:TH_*_NT) since
// the 256 MB working set exceeds the 192 MB L2 and every byte is single-use.

typedef __attribute__((ext_vector_type(4))) float v4f;

__global__ __launch_bounds__(256) void xd_perm_kernel(
    const float* __restrict__ xr,
    const float* __restrict__ xi,
    float* __restrict__ out,
    unsigned N)
{
    // Each thread handles 4 consecutive complex elements (all inside one
    // j-block, since 4 | 4096).
    unsigned n = (blockIdx.x * 256u + threadIdx.x) << 2;   // element index
    if (n >= N) return;

    // n = ((a*8 + j) << 12) + b ;  swap j -> (8-j) & 7
    unsigned j  = (n >> 12) & 7u;
    unsigned jj = (8u - j) & 7u;
    unsigned m  = (n & ~0x7000u) | (jj << 12);             // source index

    // Non-temporal 128-bit loads (gfx1250 th:TH_LOAD_NT): single-use stream.
    v4f r = __builtin_nontemporal_load((const v4f*)(xr + m));
    v4f i = __builtin_nontemporal_load((const v4f*)(xi + m));

    // Interleave into complex64 layout in registers.
    v4f o0 = { r.x, i.x, r.y, i.y };
    v4f o1 = { r.z, i.z, r.w, i.w };

    // Non-temporal 128-bit stores (th:TH_STORE_NT): don't thrash L2.
    v4f* dst = (v4f*)(out + 2u * n);                       // 32-byte aligned
    __builtin_nontemporal_store(o0, dst);
    __builtin_nontemporal_store(o1, dst + 1);
}

extern "C" void kernel_launch(void* const* d_in, const int* in_sizes, int n_in,
                              void* d_out, int out_size, void* d_ws, size_t ws_size,
                              hipStream_t stream)
{
    const float* xr = (const float*)d_in[0];   // x_real, float32, N elements
    const float* xi = (const float*)d_in[1];   // x_imag, float32, N elements
    float* out = (float*)d_out;                // complex64 viewed as 2N floats

    unsigned N = (unsigned)in_sizes[0];        // 16,777,216
    unsigned threads = (N + 3u) >> 2;          // 4 elements per thread
    dim3 block(256);
    dim3 grid((threads + 255u) / 256u);        // 16,384 blocks of 8 wave32s

    xd_perm_kernel<<<grid, block, 0, stream>>>(xr, xi, out, N);
}